// DeformRoIPooling_3882650436007
// MI455X (gfx1250) — compile-verified
//
#include <hip/hip_runtime.h>
#include <cstdint>

#define PP          7
#define NBINS       49            // 7*7
#define OC          256
#define HW          128
#define OUT_PER_ROI (OC * NBINS)  // 12544
#define META_STRIDE 21            // 4 coord dwords + 16 weights + 1 count; coprime with 64 banks
#define SLICES      4             // gridDim.y work split for latency hiding

typedef float v2f __attribute__((ext_vector_type(2), aligned(4)));

__global__ __launch_bounds__(256)
void dpsroi_kernel(const float* __restrict__ data,
                   const float* __restrict__ rois,
                   const float* __restrict__ offset,
                   float* __restrict__ out)
{
    __shared__ float s_off[2 * NBINS];            // staged deform offsets (98 floats)
    __shared__ float s_meta[NBINS * META_STRIDE]; // per-bin sample metadata

    const int n     = blockIdx.x;
    const int slice = blockIdx.y;
    const int t     = threadIdx.x;

    const float* goff = offset + (size_t)n * (2 * NBINS);

#if defined(__gfx1250__)
    // CDNA5 async global -> LDS copy of this roi's deform offsets (ASYNCcnt path).
    if (t < 2 * NBINS) {
        unsigned lds_addr = (unsigned)(unsigned long long)(void*)&s_off[t];
        asm volatile("global_load_async_to_lds_b32 %0, %1, %2"
                     :: "v"(lds_addr), "v"(t * 4), "s"(goff)
                     : "memory");
    }
    asm volatile("s_wait_asynccnt 0" ::: "memory");
#else
    if (t < 2 * NBINS) s_off[t] = goff[t];
#endif
    __syncthreads();

    // ---- ROI geometry: uniform across the block -> scalar loads / SALU ----
    const float r0 = rois[n * 5 + 0];
    const float r1 = rois[n * 5 + 1];
    const float r2 = rois[n * 5 + 2];
    const float r3 = rois[n * 5 + 3];
    const float r4 = rois[n * 5 + 4];
    const int   bidx   = (int)r0;
    const float roi_sw = rintf(r1) * 0.0625f - 0.5f;           // round-half-even == jnp.round
    const float roi_sh = rintf(r2) * 0.0625f - 0.5f;
    const float roi_ew = (rintf(r3) + 1.0f) * 0.0625f - 0.5f;
    const float roi_eh = (rintf(r4) + 1.0f) * 0.0625f - 0.5f;
    const float roi_w  = fmaxf(roi_ew - roi_sw, 0.1f);
    const float roi_h  = fmaxf(roi_eh - roi_sh, 0.1f);
    const float bin_w  = roi_w / 7.0f;
    const float bin_h  = roi_h / 7.0f;
    const float sub_w  = bin_w * 0.5f;
    const float sub_h  = bin_h * 0.5f;

    // ---- Per-bin sample metadata, computed once per block ----
    if (t < NBINS) {
        const int ph = t / PP;
        const int pw = t - ph * PP;
        const int part_h = (int)floorf((float)ph / 7.0f * 7.0f); // == ph, kept bit-faithful
        const int part_w = (int)floorf((float)pw / 7.0f * 7.0f);
        const float tx = s_off[part_h * PP + part_w] * 0.1f;
        const float ty = s_off[NBINS + part_h * PP + part_w] * 0.1f;
        const float wstart = (float)pw * bin_w + roi_sw + tx * roi_w;
        const float hstart = (float)ph * bin_h + roi_sh + ty * roi_h;

        float* m = &s_meta[t * META_STRIDE];
        int count = 0;
        #pragma unroll
        for (int s = 0; s < 4; ++s) {
            const int   ihs = s >> 1;
            const int   iws = s & 1;
            const float w = wstart + (float)iws * sub_w;
            const float h = hstart + (float)ihs * sub_h;
            const bool valid = (w >= -0.5f) && (w <= 127.5f) &&
                               (h >= -0.5f) && (h <= 127.5f);
            const float wc = fminf(fmaxf(w, 0.0f), 127.0f);
            const float hc = fminf(fmaxf(h, 0.0f), 127.0f);
            const int x1 = (int)floorf(wc);
            const int y1 = (int)floorf(hc);
            const int y2 = min(y1 + 1, HW - 1);
            const float dx = wc - (float)x1;
            const float dy = hc - (float)y1;
            const float vf = valid ? 1.0f : 0.0f;
            count += valid ? 1 : 0;

            const float w11 = (1.0f - dx) * (1.0f - dy) * vf;   // (y1,x1)
            const float w12 = (1.0f - dx) * dy * vf;            // (y2,x1)
            const float w21 = dx * (1.0f - dy) * vf;            // (y1,x2)
            const float w22 = dx * dy * vf;                     // (y2,x2)

            // b64 pair basis: load (lo,hi) at xb = min(x1,126).
            // x1 < 127 : lo = v(x1), hi = v(x1+1) = v(x2)        -> (wlo,whi) = (w11,w21)
            // x1 == 127: dx = 0 so w21 = w22 = 0, hi = v(127)    -> (wlo,whi) = (0,  w11)
            const int  xb   = min(x1, HW - 2);
            const bool edge = (x1 == HW - 1);
            m[s] = __int_as_float(xb | (y1 << 8) | (y2 << 16));
            m[4 + 4 * s + 0] = edge ? 0.0f : w11;   // * lo(row y1)
            m[4 + 4 * s + 1] = edge ? 0.0f : w12;   // * lo(row y2)
            m[4 + 4 * s + 2] = edge ? w11  : w21;   // * hi(row y1)
            m[4 + 4 * s + 3] = edge ? w12  : w22;   // * hi(row y2)
        }
        m[20] = (float)count;
    }
    __syncthreads();

    const float* dplane0 = data + ((size_t)bidx << 22);   // bidx * 256*128*128
    float*       outp    = out + (size_t)n * OUT_PER_ROI;

    // Each block handles bins k = slice, slice+SLICES, ... ; stores stay coalesced.
    #pragma unroll 2
    for (int k = slice; k < NBINS; k += SLICES) {
        const int      idx = (k << 8) + t;                 // 0 .. 12543
        const unsigned c   = (unsigned)idx / 49u;
        const unsigned bin = (unsigned)idx - c * 49u;
        const float* dp = dplane0 + ((size_t)c << 14);     // channel plane (16384 f32)
        const float* m  = &s_meta[bin * META_STRIDE];

        float acc = 0.0f;
        #pragma unroll
        for (int s = 0; s < 4; ++s) {
            const int pk = __float_as_int(m[s]);
            const int xb =  pk        & 0xff;
            const int y1 = (pk >> 8)  & 0xff;
            const int y2 = (pk >> 16) & 0xff;
            const v2f p1 = *(const v2f*)(dp + (y1 << 7) + xb);   // global_load_b64
            const v2f p2 = *(const v2f*)(dp + (y2 << 7) + xb);   // global_load_b64
            // Reference order: w11*v11 + w12*v12 + w21*v21 + w22*v22
            acc += m[4 + 4 * s + 0] * p1.x + m[4 + 4 * s + 1] * p2.x
                 + m[4 + 4 * s + 2] * p1.y + m[4 + 4 * s + 3] * p2.y;
        }
        const float cnt = m[20];
        outp[idx] = (cnt > 0.0f) ? acc / cnt : 0.0f;       // same rounding as reference
    }
}

extern "C" void kernel_launch(void* const* d_in, const int* in_sizes, int n_in,
                              void* d_out, int out_size, void* d_ws, size_t ws_size,
                              hipStream_t stream) {
    (void)n_in; (void)out_size; (void)d_ws; (void)ws_size;
    const float* data   = (const float*)d_in[0];
    const float* rois   = (const float*)d_in[1];
    const float* offset = (const float*)d_in[2];
    float* out = (float*)d_out;
    const int n_rois = in_sizes[1] / 5;   // 512
    dim3 grid(n_rois, SLICES);
    dpsroi_kernel<<<grid, 256, 0, stream>>>(data, rois, offset, out);
}